// FPNRelation_51951924413006
// MI455X (gfx1250) — compile-verified
//
#include <hip/hip_runtime.h>

typedef __attribute__((ext_vector_type(2))) float v2f;
typedef __attribute__((ext_vector_type(8))) float v8f;

#define CC 256
#define HW 1024  // 32*32

// ---------------- Kernel 1: 4x4 average pool: x0 [8,256,128,128] -> pooled [8,256,32,32]
__global__ void pool4_kernel(const float* __restrict__ x0, float* __restrict__ pooled) {
  int i = blockIdx.x * blockDim.x + threadIdx.x;  // exactly 8*256*1024 threads
  int px = i & 31;
  int py = (i >> 5) & 31;
  int bc = i >> 10;
  const float* src = x0 + ((size_t)bc << 14) + (size_t)py * 512 + px * 4;
  float s = 0.f;
#pragma unroll
  for (int r = 0; r < 4; ++r) {
    float4 v = *(const float4*)(src + (size_t)r * 128);
    s += (v.x + v.y) + (v.z + v.w);
  }
  pooled[i] = s * 0.0625f;
}

// ---------------- Kernel 1b: repack 3x3 weights [c*9+tap] -> tap-major [tap*256+c]
__global__ void repack3x3_kernel(const float* __restrict__ w0, const float* __restrict__ w1,
                                 float* __restrict__ w0t, float* __restrict__ w1t) {
  int i = blockIdx.x * blockDim.x + threadIdx.x;  // 2304
  if (i >= CC * 9) return;
  int c = i / 9, tap = i % 9;
  w0t[tap * CC + c] = w0[i];
  w1t[tap * CC + c] = w1[i];
}

// ---------------- Kernel 2: two 1x1 convs via V_WMMA_F32_16X16X4_F32
// Per wave: 16 positions (one batch), K-loop over 256 channels in steps of 4.
// A (16x4 fp32): lanes 0-15 hold K={k0,k0+1} for M=lane; lanes 16-31 hold K={k0+2,k0+3}.
// B (4x16): VGPR0 = rows {K=k0 (lanes 0-15), K=k0+2 (lanes 16-31)}, VGPR1 = {k0+1, k0+3}.
// Only columns N=0 (w0) and N=1 (w1) are non-zero -> built branchlessly with 0/1 selects.
__global__ void __launch_bounds__(32) conv1x1_wmma_kernel(
    const float* __restrict__ pooled,
    const float* __restrict__ w0, const float* __restrict__ w1,
    const float* __restrict__ b0p, const float* __restrict__ b1p,
    float* __restrict__ m1, float* __restrict__ m2) {
  int b = blockIdx.x >> 6;    // batch
  int t = blockIdx.x & 63;    // 16-position tile
  int lane = (int)threadIdx.x;
  int hl = lane >> 4;         // 0 -> K pair {k0,k0+1}, 1 -> {k0+2,k0+3}
  int lm = lane & 15;
  int p = t * 16 + lm;        // A row (position); also D column index
  const float* pa = pooled + (size_t)b * (CC * HW) + (size_t)(2 * hl) * HW + p;
  float sel0 = (lm == 0) ? 1.f : 0.f;
  float sel1 = (lm == 1) ? 1.f : 0.f;

  v8f acc = {};
  for (int k0 = 0; k0 < CC; k0 += 4) {
    v2f A;
    A.x = pa[(size_t)k0 * HW];
    A.y = pa[(size_t)k0 * HW + HW];
    // wave-uniform addresses -> scalar loads
    float4 q0 = *(const float4*)(w0 + k0);
    float4 q1 = *(const float4*)(w1 + k0);
    float w0lo = hl ? q0.z : q0.x, w0hi = hl ? q0.w : q0.y;
    float w1lo = hl ? q1.z : q1.x, w1hi = hl ? q1.w : q1.y;
    v2f Bv;
    Bv.x = sel0 * w0lo + sel1 * w1lo;
    Bv.y = sel0 * w0hi + sel1 * w1hi;
    acc = __builtin_amdgcn_wmma_f32_16x16x4_f32(false, A, false, Bv, (short)0, acc,
                                                false, false);
  }
  // D layout: lane lm = column N; VGPR v holds row M = 8*hl + v.
  float bias = (lm == 0) ? b0p[0] : b1p[0];
  float* dst = (lm == 0) ? m1 : m2;
  if (lm < 2) {
#pragma unroll
    for (int v = 0; v < 8; ++v) {
      int row = 8 * hl + v;
      dst[(size_t)b * HW + t * 16 + row] = fmaxf(acc[v] + bias, 0.f);
    }
  }
}

// ---------------- Kernel 3: two 3x3 convs (pad=1) on x2 via WMMA, im2col over 9 taps
// Weights pre-repacked to tap-major [tap*256+c] so the inner loop does one uniform
// float4 (scalar) load per weight tensor.
__global__ void __launch_bounds__(32) conv3x3_wmma_kernel(
    const float* __restrict__ xin,                               // x2 [8,256,32,32]
    const float* __restrict__ w0t, const float* __restrict__ w1t,// [9][256]
    const float* __restrict__ b0p, const float* __restrict__ b1p,
    float* __restrict__ c1, float* __restrict__ c2) {
  int b = blockIdx.x >> 6;
  int t = blockIdx.x & 63;
  int lane = (int)threadIdx.x;
  int hl = lane >> 4;
  int lm = lane & 15;
  int p = t * 16 + lm;
  int y = p >> 5, x = p & 31;
  const float* pb = xin + (size_t)b * (CC * HW) + (size_t)(2 * hl) * HW;
  float sel0 = (lm == 0) ? 1.f : 0.f;
  float sel1 = (lm == 1) ? 1.f : 0.f;

  v8f acc = {};
  for (int tap = 0; tap < 9; ++tap) {
    int dy = tap / 3 - 1, dx = tap % 3 - 1;
    int yy = y + dy, xx = x + dx;
    bool valid = (yy >= 0) && (yy < 32) && (xx >= 0) && (xx < 32);
    int yc = min(max(yy, 0), 31), xc = min(max(xx, 0), 31);
    float msk = valid ? 1.f : 0.f;
    const float* pa = pb + yc * 32 + xc;
    const float* wt0 = w0t + tap * CC;
    const float* wt1 = w1t + tap * CC;
    for (int k0 = 0; k0 < CC; k0 += 4) {
      v2f A;
      A.x = msk * pa[(size_t)k0 * HW];
      A.y = msk * pa[(size_t)k0 * HW + HW];
      float4 q0 = *(const float4*)(wt0 + k0);   // uniform -> scalar load
      float4 q1 = *(const float4*)(wt1 + k0);
      float w0lo = hl ? q0.z : q0.x, w0hi = hl ? q0.w : q0.y;
      float w1lo = hl ? q1.z : q1.x, w1hi = hl ? q1.w : q1.y;
      v2f Bv;
      Bv.x = sel0 * w0lo + sel1 * w1lo;
      Bv.y = sel0 * w0hi + sel1 * w1hi;
      acc = __builtin_amdgcn_wmma_f32_16x16x4_f32(false, A, false, Bv, (short)0, acc,
                                                  false, false);
    }
  }
  float bias = (lm == 0) ? b0p[0] : b1p[0];
  float* dst = (lm == 0) ? c1 : c2;
  if (lm < 2) {
#pragma unroll
    for (int v = 0; v < 8; ++v) {
      int row = 8 * hl + v;
      dst[(size_t)b * HW + t * 16 + row] = fmaxf(acc[v] + bias, 0.f);
    }
  }
}

// ---------------- Kernel 4: per-batch means + fused 32x32 map
// map32 = m1 * mean(m2) + c1 * mean(c2)   (both maps share the same nearest-resize)
__global__ void combine_kernel(const float* __restrict__ m1, const float* __restrict__ m2,
                               const float* __restrict__ c1, const float* __restrict__ c2,
                               float* __restrict__ map32) {
  int b = blockIdx.x;
  int tid = (int)threadIdx.x;  // 256 threads
  __shared__ float s2[256], sc[256];
  float a2 = 0.f, ac = 0.f;
  for (int i = tid; i < HW; i += 256) {
    a2 += m2[(size_t)b * HW + i];
    ac += c2[(size_t)b * HW + i];
  }
  s2[tid] = a2; sc[tid] = ac;
  __syncthreads();
  for (int s = 128; s > 0; s >>= 1) {
    if (tid < s) { s2[tid] += s2[tid + s]; sc[tid] += sc[tid + s]; }
    __syncthreads();
  }
  float mean2 = s2[0] * (1.f / 1024.f);
  float meanc = sc[0] * (1.f / 1024.f);
  for (int i = tid; i < HW; i += 256) {
    map32[(size_t)b * HW + i] =
        m1[(size_t)b * HW + i] * mean2 + c1[(size_t)b * HW + i] * meanc;
  }
}

// ---------------- Kernel 5: out = x + nearest_resize(map32)  (float4 streamed)
// nearest index for target S=2^lg from 32: (i*32)/S == (i<<5)>>lg (pure shifts)
__global__ void addmap_kernel(const float* __restrict__ xin,
                              const float* __restrict__ map32,
                              float* __restrict__ out, int lg, long n4) {
  long i = blockIdx.x * (long)blockDim.x + threadIdx.x;
  if (i >= n4) return;
  long e = i << 2;
  int S = 1 << lg;
  int sp = (int)(e & ((1L << (2 * lg)) - 1));  // spatial index in channel plane
  int b = (int)(e >> (8 + 2 * lg));            // / (256*S*S)
  int y = sp >> lg;
  int x = sp & (S - 1);
  const float* mrow = map32 + (size_t)b * HW + ((size_t)((y << 5) >> lg) << 5);
  float4 v = *(const float4*)(xin + e);
  float4 o;
  o.x = v.x + mrow[((x + 0) << 5) >> lg];
  o.y = v.y + mrow[((x + 1) << 5) >> lg];
  o.z = v.z + mrow[((x + 2) << 5) >> lg];
  o.w = v.w + mrow[((x + 3) << 5) >> lg];
  *(float4*)(out + e) = o;
}

extern "C" void kernel_launch(void* const* d_in, const int* in_sizes, int n_in,
                              void* d_out, int out_size, void* d_ws, size_t ws_size,
                              hipStream_t stream) {
  (void)in_sizes; (void)n_in; (void)out_size; (void)ws_size;
  const float* x0 = (const float*)d_in[0];
  const float* x1 = (const float*)d_in[1];
  const float* x2 = (const float*)d_in[2];
  const float* x3 = (const float*)d_in[3];
  const float* x4 = (const float*)d_in[4];
  const float* com_w0 = (const float*)d_in[5];
  const float* com_b0 = (const float*)d_in[6];
  const float* com_w1 = (const float*)d_in[7];
  const float* com_b1 = (const float*)d_in[8];
  const float* en_w0 = (const float*)d_in[9];
  const float* en_b0 = (const float*)d_in[10];
  const float* en_w1 = (const float*)d_in[11];
  const float* en_b1 = (const float*)d_in[12];

  // Workspace layout (floats); every buffer is fully written before being read.
  float* ws = (float*)d_ws;
  float* pooled = ws;               // 8*256*1024 = 2,097,152
  float* m1 = pooled + 2097152;     // 8*1024 each
  float* m2 = m1 + 8192;
  float* c1 = m2 + 8192;
  float* c2 = c1 + 8192;
  float* map32 = c2 + 8192;
  float* w0t = map32 + 8192;        // 9*256 (16B aligned: offset multiple of 4)
  float* w1t = w0t + 2304;
  // total ~8.6 MB

  pool4_kernel<<<8192, 256, 0, stream>>>(x0, pooled);
  repack3x3_kernel<<<9, 256, 0, stream>>>(en_w0, en_w1, w0t, w1t);
  conv1x1_wmma_kernel<<<512, 32, 0, stream>>>(pooled, com_w0, com_w1, com_b0, com_b1, m1, m2);
  conv3x3_wmma_kernel<<<512, 32, 0, stream>>>(x2, w0t, w1t, en_b0, en_b1, c1, c2);
  combine_kernel<<<8, 256, 0, stream>>>(m1, m2, c1, c2, map32);

  float* out = (float*)d_out;
  const float* xs[5] = {x0, x1, x2, x3, x4};
  const int lgs[5] = {7, 6, 5, 4, 3};
  size_t off = 0;
  for (int l = 0; l < 5; ++l) {
    long S = 1L << lgs[l];
    long n = 8L * 256L * S * S;
    long n4 = n >> 2;
    int blocks = (int)((n4 + 255) / 256);
    addmap_kernel<<<blocks, 256, 0, stream>>>(xs[l], map32, out + off, lgs[l], n4);
    off += (size_t)n;
  }
}